// MultiVariateNormal_54262616818087
// MI455X (gfx1250) — compile-verified
//
#include <hip/hip_runtime.h>

// out_n = exp(-0.5 * d^T Sigma^-1 d) / sqrt(4*pi^2*det(Sigma)),  d = X_n - mu
//
// Memory-bound streaming kernel (12 B/point, ~201 MB @ 23.3 TB/s ~ 8.6 us).
// Quadratic form computed on the CDNA5 matrix path via V_WMMA_F32_16X16X4_F32:
//   A(16x4) = weights (ia, 2*ib, id, 0) broadcast over rows  -> D[m,n] = y_n for all m,
//   B(4x16) = per-point features (dx^2, dx*dy, dy^2, 0) as columns,
// so each lane L ends up with y_{L&15} replicated in all 8 C VGPRs: no result
// extraction needed. Two WMMAs per wave cover 32 points (one per lane).

typedef __attribute__((ext_vector_type(2))) float v2f;
typedef __attribute__((ext_vector_type(8))) float v8f;

// lane L <-> lane L^16 exchange (group-of-32 swizzle: and=0x1f, or=0, xor=0x10)
__device__ __forceinline__ float swz_xor16(float x) {
  return __int_as_float(__builtin_amdgcn_ds_swizzle(__float_as_int(x), 0x401f));
}

__global__ __launch_bounds__(256) void mvn_wmma_kernel(
    const float* __restrict__ X, const float* __restrict__ mu,
    const float* __restrict__ cov, float* __restrict__ out) {
  const int i    = blockIdx.x * 256 + threadIdx.x;   // one point per thread
  const int lane = threadIdx.x & 31;
  const bool lo  = lane < 16;

  // Uniform 2x2 inverse / det / normalization (scalarized by the compiler).
  const float a = cov[0], b = cov[1], c = cov[2], d = cov[3];
  const float det  = a * d - b * c;
  const float rdet = 1.0f / det;
  const float ia   = d * rdet;          // inv[0][0]
  const float ib2  = -(b + c) * rdet;   // inv[0][1] + inv[1][0]
  const float id_  = a * rdet;          // inv[1][1]
  const float invnorm = 0.15915494309189535f * __frsqrt_rn(det); // 1/(2*pi*sqrt(det))

  // Coalesced float2 load: wave reads 256 contiguous bytes.
  const v2f p = ((const v2f*)X)[i];
  const float dx = p.x - mu[0];
  const float dy = p.y - mu[1];

  // Partner half-wave's diffs (point i^16 within the wave's 32-point chunk).
  const float sx = swz_xor16(dx);
  const float sy = swz_xor16(dy);

  // A = weight matrix, identical in every row:
  //   lanes 0-15 supply K=0 (VGPR0) and K=1 (VGPR1); lanes 16-31 supply K=2, K=3.
  v2f A;
  A.x = lo ? ia  : id_;
  A.y = lo ? ib2 : 0.0f;

  // B tile0 = features of points [base, base+16):
  //   lanes 0-15: rows K=0/1 of their own point; lanes 16-31: rows K=2/3 of point lane-16.
  v2f B0;
  B0.x = lo ? dx * dx : sy * sy;
  B0.y = lo ? dx * dy : 0.0f;

  // B tile1 = features of points [base+16, base+32): swap local/swizzled roles.
  v2f B1;
  B1.x = lo ? sx * sx : dy * dy;
  B1.y = lo ? sx * sy : 0.0f;

  const v8f C = {};
  // (neg_a, A, neg_b, B, c_mod, C, reuse_a, reuse_b)
  v8f D0 = __builtin_amdgcn_wmma_f32_16x16x4_f32(false, A, false, B0, (short)0, C, false, false);
  v8f D1 = __builtin_amdgcn_wmma_f32_16x16x4_f32(false, A, false, B1, (short)0, C, false, false);

  // Lane L holds y_{base + (L&15)} (tile0) / y_{base+16+(L&15)} (tile1) in every
  // C VGPR; picking per-half gives lane L exactly y_{base+L}.
  const float md = lo ? D0[0] : D1[0];
  out[i] = __expf(-0.5f * md) * invnorm;   // coalesced 128 B/wave store
}

// Plain-VALU tail for n not a multiple of 256 (keeps WMMA kernel EXEC-all-ones).
__global__ void mvn_tail_kernel(const float* __restrict__ X,
                                const float* __restrict__ mu,
                                const float* __restrict__ cov,
                                float* __restrict__ out, int start, int n) {
  const int i = start + blockIdx.x * blockDim.x + threadIdx.x;
  if (i >= n) return;
  const float a = cov[0], b = cov[1], c = cov[2], d = cov[3];
  const float det  = a * d - b * c;
  const float rdet = 1.0f / det;
  const float dx = X[2 * i + 0] - mu[0];
  const float dy = X[2 * i + 1] - mu[1];
  const float md = (d * dx * dx - (b + c) * dx * dy + a * dy * dy) * rdet;
  out[i] = __expf(-0.5f * md) * (0.15915494309189535f * __frsqrt_rn(det));
}

extern "C" void kernel_launch(void* const* d_in, const int* in_sizes, int n_in,
                              void* d_out, int out_size, void* d_ws, size_t ws_size,
                              hipStream_t stream) {
  const float* X   = (const float*)d_in[0];   // (n, 2) float32
  const float* mu  = (const float*)d_in[1];   // (2,)   float32
  const float* cov = (const float*)d_in[2];   // (2, 2) float32
  float* out = (float*)d_out;                 // (n,)   float32

  const int n     = out_size;                 // number of points
  const int nmain = n & ~255;                 // multiple of 256 -> full waves
  if (nmain > 0) {
    mvn_wmma_kernel<<<nmain / 256, 256, 0, stream>>>(X, mu, cov, out);
  }
  const int ntail = n - nmain;
  if (ntail > 0) {
    mvn_tail_kernel<<<(ntail + 255) / 256, 256, 0, stream>>>(X, mu, cov, out, nmain, n);
  }
}